// MLIF_29326036697275
// MI455X (gfx1250) — compile-verified
//
#include <hip/hip_runtime.h>
#include <math.h>

typedef float v4f __attribute__((ext_vector_type(4)));

// Exact pointee type expected by the gfx1250 async-copy builtins
// (diagnostic: "__attribute__((__vector_size__(4 * sizeof(int)))) int __device__ *").
typedef int vi4 __attribute__((vector_size(16)));
typedef __attribute__((address_space(1))) vi4 as1_vi4;   // global (prints as __device__)
typedef __attribute__((address_space(3))) vi4 as3_vi4;   // LDS    (prints as __shared__)

#define NSPACE  (128u * 65536u)   // floats per time plane = 8,388,608
#define NVEC    (NSPACE / 4u)     // float4 per time plane = 2,097,152
#define TSTEPS  4
#define BLOCK   256u              // 8 wave32 per workgroup

#if defined(__AMDGCN__) &&                                                     \
    __has_builtin(__builtin_amdgcn_global_load_async_to_lds_b128) &&           \
    __has_builtin(__builtin_amdgcn_s_wait_asynccnt)
#define USE_ASYNC_LDS 1
#else
#define USE_ASYNC_LDS 0
#endif

__global__ __launch_bounds__(BLOCK) void mlif_scan_kernel(
    const v4f* __restrict__ x,      // (T, NVEC) float4
    const float* __restrict__ S,    // scalar parameter
    v4f* __restrict__ spikes,       // (T, NVEC) float4
    v4f* __restrict__ mems)         // (T, NVEC) float4
{
    const unsigned i = blockIdx.x * BLOCK + threadIdx.x;   // vec4 index in a plane
    const float sig = 1.0f / (1.0f + expf(-S[0]));         // sigmoid(S), wave-uniform

    v4f vmem = {0.0f, 0.0f, 0.0f, 0.0f};

#if USE_ASYNC_LDS
    // Double-buffered async staging of x time-planes through LDS (ASYNCcnt pipeline).
    __shared__ v4f tile[2][BLOCK];
    __builtin_amdgcn_global_load_async_to_lds_b128(
        (as1_vi4*)(void*)(x + i),
        (as3_vi4*)&tile[0][threadIdx.x],
        0, 0);
#endif

#pragma unroll
    for (int t = 0; t < TSTEPS; ++t) {
        const unsigned idx = (unsigned)t * NVEC + i;
        v4f xt;
#if USE_ASYNC_LDS
        if (t + 1 < TSTEPS) {
            __builtin_amdgcn_global_load_async_to_lds_b128(
                (as1_vi4*)(void*)(x + (idx + NVEC)),
                (as3_vi4*)&tile[(t + 1) & 1][threadIdx.x],
                0, 0);
            __builtin_amdgcn_s_wait_asynccnt(1);   // plane t landed; t+1 in flight
        } else {
            __builtin_amdgcn_s_wait_asynccnt(0);
        }
        xt = tile[t & 1][threadIdx.x];
#else
        // Streaming (non-temporal) b128 load; prefetch next plane into GL2.
        xt = __builtin_nontemporal_load(x + idx);
        if (t + 1 < TSTEPS)
            __builtin_prefetch((const void*)(x + (idx + NVEC)), 0, 0);
#endif

        v4f sp, mm;
#pragma unroll
        for (int k = 0; k < 4; ++k) {
            float v  = (vmem[k] + xt[k]) * sig;            // integrate + learned scale
            float q  = __builtin_rintf(v * 65536.0f);      // round-half-even (v_rndne)
            q        = fminf(fmaxf(q, 0.0f), 65536.0f);    // clamp(round(z), 0, L)
            float spike = q * (1.0f / 65536.0f);           // graded spike in [0,1]
            float vr    = (q > 0.0f) ? 0.0f : v;           // hard reset (fired = q>0)
            sp[k]   = spike;
            mm[k]   = vr;                                  // mems output: post-reset v
            vmem[k] = vr * 0.5f;                           // DELTA-decayed carried state
        }

        // Outputs are write-once streams: non-temporal stores keep L2 clean.
        __builtin_nontemporal_store(sp, spikes + idx);
        __builtin_nontemporal_store(mm, mems + idx);
    }
}

extern "C" void kernel_launch(void* const* d_in, const int* in_sizes, int n_in,
                              void* d_out, int out_size, void* d_ws, size_t ws_size,
                              hipStream_t stream) {
    (void)in_sizes; (void)n_in; (void)out_size; (void)d_ws; (void)ws_size;

    const v4f*   x = (const v4f*)d_in[0];    // (4, 128, 65536) f32
    const float* S = (const float*)d_in[1];  // scalar

    float* out    = (float*)d_out;
    v4f*   spikes = (v4f*)out;                            // first 4*NSPACE floats
    v4f*   mems   = (v4f*)(out + 4ull * (size_t)NSPACE);  // next 4*NSPACE floats

    mlif_scan_kernel<<<NVEC / BLOCK, BLOCK, 0, stream>>>(x, S, spikes, mems);
}